// GraphEncoder_71347996721729
// MI455X (gfx1250) — compile-verified
//
#include <hip/hip_runtime.h>

typedef __attribute__((ext_vector_type(2))) float v2f;
typedef __attribute__((ext_vector_type(8))) float v8f;
typedef __attribute__((ext_vector_type(4))) int   v4i;

#define AS1 __attribute__((address_space(1)))
#define AS3 __attribute__((address_space(3)))

#define NN 50000
#define NE 800000
#define NG 512
#define CH 256
#define BN_EPS 1e-5f

#if defined(__AMDGCN__) && __has_builtin(__builtin_amdgcn_global_load_async_to_lds_b128)
#define HAVE_ASYNC_LDS 1
#else
#define HAVE_ASYNC_LDS 0
#endif

__device__ __forceinline__ void wait_async_lds() {
#if __has_builtin(__builtin_amdgcn_s_wait_asynccnt)
    __builtin_amdgcn_s_wait_asynccnt(0);
#elif defined(__AMDGCN__)
    asm volatile("s_wait_asynccnt 0x0" ::: "memory");
#endif
}

// ---------------- utility ----------------
__global__ void zero_kernel(float* __restrict__ p, long n) {
    long i = (long)blockIdx.x * blockDim.x + threadIdx.x;
    if (i < n) p[i] = 0.0f;
}

__global__ void degree_kernel(const int* __restrict__ dst, float* __restrict__ deg, int E) {
    int e = blockIdx.x * blockDim.x + threadIdx.x;
    if (e < E) atomicAdd(&deg[dst[e]], 1.0f);
}

__global__ void dinv_kernel(float* __restrict__ deg, int n) {
    int i = blockIdx.x * blockDim.x + threadIdx.x;
    if (i < n) deg[i] = rsqrtf(deg[i] + 1.0f);
}

// ---------------- fp32 WMMA GEMM: out[M,N] = A[M,K] @ W[K,N] ----------------
// Requirements: M % 16 == 0, K % 4 == 0 (K <= 256), N == 256.
// Block = 128 threads (4 waves) computing a 16x256 output tile:
//   - 16xK A tile staged once into LDS (async global->LDS when available)
//   - wave w owns columns [64w, 64w+64): 4 accumulators, A fragment reused 4x
// mode: 0 = raw store, 1 = +bias, 2 = relu(+bias)
__global__ void wmma_gemm_f32(const float* __restrict__ A,
                              const float* __restrict__ W,
                              const float* __restrict__ bias,
                              float* __restrict__ out,
                              int M, int K, int N, int mode) {
    __shared__ float sA[16 * 256];   // 16 KB max (16 rows x K<=256)

    const int m0 = blockIdx.x * 16;

    // ---- stage A tile (rows m0..m0+15 are contiguous: 16*K floats) ----
    const float* gsrc = A + (long)m0 * K;
    const int nquads = 4 * K;        // (16*K)/4 float4 elements
#if HAVE_ASYNC_LDS
    for (int i = threadIdx.x; i < nquads; i += blockDim.x) {
        __builtin_amdgcn_global_load_async_to_lds_b128(
            (AS1 v4i*)(gsrc + 4 * i),
            (AS3 v4i*)(sA + 4 * i),
            0, 0);
    }
    wait_async_lds();
    __syncthreads();
#else
    for (int i = threadIdx.x; i < nquads; i += blockDim.x) {
        ((float4*)sA)[i] = ((const float4*)gsrc)[i];
    }
    __syncthreads();
#endif

    const int wave = threadIdx.x >> 5;
    const int lane = threadIdx.x & 31;
    const int lh   = lane & 15;   // row (A) / column-in-subtile (B,D)
    const int sel  = lane >> 4;   // 0: K pair {0,1}, 1: K pair {2,3}

    const int cbase = wave * 64;

    // A fragments from LDS: lane lh holds row lh; pairs of K at (k0/2 + sel)
    const float2* __restrict__ arow2 = (const float2*)(sA + (long)lh * K);
    // B base: row (2*sel) of W at this wave's columns
    const float* __restrict__ wb = W + (long)(2 * sel) * N + cbase + lh;

    v8f acc0 = {}, acc1 = {}, acc2 = {}, acc3 = {};
    for (int k0 = 0; k0 < K; k0 += 4) {
        float2 af = arow2[(k0 >> 1) + sel];
        v2f a;
        a.x = af.x;
        a.y = af.y;
        long r0 = (long)k0 * N;
        long r1 = r0 + N;
        v2f b0, b1, b2, b3;
        b0.x = wb[r0];      b0.y = wb[r1];
        b1.x = wb[r0 + 16]; b1.y = wb[r1 + 16];
        b2.x = wb[r0 + 32]; b2.y = wb[r1 + 32];
        b3.x = wb[r0 + 48]; b3.y = wb[r1 + 48];
        acc0 = __builtin_amdgcn_wmma_f32_16x16x4_f32(false, a, false, b0, (short)0, acc0, false, false);
        acc1 = __builtin_amdgcn_wmma_f32_16x16x4_f32(false, a, false, b1, (short)0, acc1, false, false);
        acc2 = __builtin_amdgcn_wmma_f32_16x16x4_f32(false, a, false, b2, (short)0, acc2, false, false);
        acc3 = __builtin_amdgcn_wmma_f32_16x16x4_f32(false, a, false, b3, (short)0, acc3, false, false);
    }

    // ---- epilogue / store (D layout: VGPR r -> M=r+8*sel, N=lh per subtile) ----
    v8f accs[4] = {acc0, acc1, acc2, acc3};
#pragma unroll
    for (int j = 0; j < 4; ++j) {
        int col = cbase + 16 * j + lh;
        float bv = (mode > 0) ? bias[col] : 0.0f;
#pragma unroll
        for (int r = 0; r < 8; ++r) {
            int orow = m0 + r + 8 * sel;
            float v = accs[j][r] + bv;
            if (mode == 2) v = fmaxf(v, 0.0f);
            out[(long)orow * N + col] = v;
        }
    }
}

// ---------------- edge scatter: agg[dst] += t[src] * dinv[src]*dinv[dst] ----
// 64 threads per edge, 4 channels per thread (float4 gather).
__global__ void scatter_kernel(const float* __restrict__ t,
                               const int* __restrict__ src,
                               const int* __restrict__ dst,
                               const float* __restrict__ dinv,
                               float* __restrict__ agg, int E) {
    long idx = (long)blockIdx.x * blockDim.x + threadIdx.x;
    long e = idx >> 6;
    if (e >= E) return;
    int c = (int)(idx & 63) * 4;
    int s = src[e];
    int d = dst[e];
    float w = dinv[s] * dinv[d];
    const float4 v = *(const float4*)(t + (long)s * CH + c);
    float* o = agg + (long)d * CH + c;
    atomicAdd(o + 0, v.x * w);
    atomicAdd(o + 1, v.y * w);
    atomicAdd(o + 2, v.z * w);
    atomicAdd(o + 3, v.w * w);
}

// ---------------- layer-1 epilogue: h = relu(agg + t*dinv^2 + bias) ---------
__global__ void combine_relu_kernel(const float* __restrict__ agg,
                                    const float* __restrict__ t,
                                    const float* __restrict__ dinv,
                                    const float* __restrict__ bias,
                                    float* __restrict__ h, int n) {
    long i = (long)blockIdx.x * blockDim.x + threadIdx.x;
    if (i >= (long)n * CH) return;
    int node = (int)(i >> 8);
    int c = (int)(i & (CH - 1));
    float di = dinv[node];
    float v = agg[i] + t[i] * di * di + bias[c];
    h[i] = fmaxf(v, 0.0f);
}

// ---------------- layers 2/3: y = agg + t*dinv^2 + bias (in place into agg),
// with per-channel sum / sumsq partial reduction (one atomic per channel/block)
__global__ void combine_stats_kernel(float* __restrict__ agg,
                                     const float* __restrict__ t,
                                     const float* __restrict__ dinv,
                                     const float* __restrict__ bias,
                                     float* __restrict__ sum,
                                     float* __restrict__ sumsq, int n) {
    int c  = threadIdx.x;        // 256 threads = 256 channels
    int r0 = blockIdx.x * 64;    // 64 rows per block
    float b = bias[c];
    float s = 0.0f, sq = 0.0f;
    int rend = min(r0 + 64, n);
    for (int r = r0; r < rend; ++r) {
        float di = dinv[r];
        long i = (long)r * CH + c;
        float v = agg[i] + t[i] * di * di + b;
        agg[i] = v;
        s += v;
        sq += v * v;
    }
    atomicAdd(&sum[c], s);
    atomicAdd(&sumsq[c], sq);
}

// ---------------- BatchNorm (training, population var) + ReLU ---------------
__global__ void bn_relu_kernel(const float* __restrict__ y,
                               const float* __restrict__ sum,
                               const float* __restrict__ sumsq,
                               const float* __restrict__ g,
                               const float* __restrict__ b,
                               float* __restrict__ h, int n) {
    long i = (long)blockIdx.x * blockDim.x + threadIdx.x;
    if (i >= (long)n * CH) return;
    int c = (int)(i & (CH - 1));
    float inv_n = 1.0f / (float)n;
    float mu  = sum[c] * inv_n;
    float var = sumsq[c] * inv_n - mu * mu;
    float v = (y[i] - mu) * rsqrtf(var + BN_EPS) * g[c] + b[c];
    h[i] = fmaxf(v, 0.0f);
}

// ---------------- global mean pool ------------------------------------------
__global__ void pool_accum_kernel(const float* __restrict__ h,
                                  const int* __restrict__ batch,
                                  float* __restrict__ pooled, int n) {
    long i = (long)blockIdx.x * blockDim.x + threadIdx.x;
    if (i >= (long)n * CH) return;
    int node = (int)(i >> 8);
    int c = (int)(i & (CH - 1));
    atomicAdd(&pooled[(long)batch[node] * CH + c], h[i]);
}

__global__ void pool_count_kernel(const int* __restrict__ batch,
                                  float* __restrict__ cnt, int n) {
    int i = blockIdx.x * blockDim.x + threadIdx.x;
    if (i < n) atomicAdd(&cnt[batch[i]], 1.0f);
}

__global__ void pool_div_kernel(const float* __restrict__ pooled,
                                const float* __restrict__ cnt,
                                float* __restrict__ g, int ng) {
    long i = (long)blockIdx.x * blockDim.x + threadIdx.x;
    if (i >= (long)ng * CH) return;
    int gi = (int)(i >> 8);
    g[i] = pooled[i] / fmaxf(cnt[gi], 1.0f);
}

// ---------------- orchestration ---------------------------------------------
extern "C" void kernel_launch(void* const* d_in, const int* in_sizes, int n_in,
                              void* d_out, int out_size, void* d_ws, size_t ws_size,
                              hipStream_t stream) {
    const float* x      = (const float*)d_in[0];   // [50000,128]
    const int*   ei     = (const int*)d_in[1];     // [2,800000]
    const int*   src    = ei;
    const int*   dst    = ei + NE;
    const int*   batch  = (const int*)d_in[2];     // [50000]
    const float* W1     = (const float*)d_in[3];
    const float* b1     = (const float*)d_in[4];
    const float* W2     = (const float*)d_in[5];
    const float* b2     = (const float*)d_in[6];
    const float* W3     = (const float*)d_in[7];
    const float* b3     = (const float*)d_in[8];
    const float* bn2_g  = (const float*)d_in[9];
    const float* bn2_b  = (const float*)d_in[10];
    const float* bn4_g  = (const float*)d_in[11];
    const float* bn4_b  = (const float*)d_in[12];
    const float* lin1_W = (const float*)d_in[13];
    const float* lin1_b = (const float*)d_in[14];
    const float* lin2_W = (const float*)d_in[15];
    const float* lin2_b = (const float*)d_in[16];
    float* out = (float*)d_out;                    // [512,256]

    // workspace layout (floats)
    const long FEAT = (long)NN * CH;               // 12,800,000
    float* ws     = (float*)d_ws;
    float* dinv   = ws;                            // 50000 (also deg scratch)
    float* t      = ws + 50176;                    // [NN,CH]
    float* agg    = t + FEAT;                      // [NN,CH], becomes pre-BN y
    float* h      = agg + FEAT;                    // [NN,CH]
    float* bn_sum = h + FEAT;                      // 256
    float* bn_sq  = bn_sum + CH;                   // 256
    float* pooled = bn_sq + CH;                    // [512,256]
    float* cnt    = pooled + (long)NG * CH;        // 512
    float* gbuf   = cnt + NG;                      // [512,256]
    float* g1buf  = gbuf + (long)NG * CH;          // [512,256]

    const int TB = 256;
    dim3 gemm_blk(128);                            // 4 waves
    dim3 gemm_big(NN / 16);                        // 3125 tiles of 16 rows
    dim3 gemm_mlp(NG / 16);                        // 32 tiles
    long nfe = FEAT;
    int  fe_blocks = (int)((nfe + TB - 1) / TB);
    int  sc_blocks = (int)(((long)NE * 64 + TB - 1) / TB);

    // degrees -> dinv
    zero_kernel<<<(NN + TB - 1) / TB, TB, 0, stream>>>(dinv, NN);
    degree_kernel<<<(NE + TB - 1) / TB, TB, 0, stream>>>(dst, dinv, NE);
    dinv_kernel<<<(NN + TB - 1) / TB, TB, 0, stream>>>(dinv, NN);

    // ---- layer 1: h = relu(gcn(x, W1, b1)) ----
    wmma_gemm_f32<<<gemm_big, gemm_blk, 0, stream>>>(x, W1, nullptr, t, NN, 128, CH, 0);
    zero_kernel<<<fe_blocks, TB, 0, stream>>>(agg, nfe);
    scatter_kernel<<<sc_blocks, TB, 0, stream>>>(t, src, dst, dinv, agg, NE);
    combine_relu_kernel<<<fe_blocks, TB, 0, stream>>>(agg, t, dinv, b1, h, NN);

    // ---- layer 2: h = relu(BN(gcn(h, W2, b2))) ----
    wmma_gemm_f32<<<gemm_big, gemm_blk, 0, stream>>>(h, W2, nullptr, t, NN, CH, CH, 0);
    zero_kernel<<<fe_blocks, TB, 0, stream>>>(agg, nfe);
    zero_kernel<<<2, TB, 0, stream>>>(bn_sum, 2 * CH);
    scatter_kernel<<<sc_blocks, TB, 0, stream>>>(t, src, dst, dinv, agg, NE);
    combine_stats_kernel<<<(NN + 63) / 64, CH, 0, stream>>>(agg, t, dinv, b2, bn_sum, bn_sq, NN);
    bn_relu_kernel<<<fe_blocks, TB, 0, stream>>>(agg, bn_sum, bn_sq, bn2_g, bn2_b, h, NN);

    // ---- layer 3: h = relu(BN(gcn(h, W3, b3))) ----
    wmma_gemm_f32<<<gemm_big, gemm_blk, 0, stream>>>(h, W3, nullptr, t, NN, CH, CH, 0);
    zero_kernel<<<fe_blocks, TB, 0, stream>>>(agg, nfe);
    zero_kernel<<<2, TB, 0, stream>>>(bn_sum, 2 * CH);
    scatter_kernel<<<sc_blocks, TB, 0, stream>>>(t, src, dst, dinv, agg, NE);
    combine_stats_kernel<<<(NN + 63) / 64, CH, 0, stream>>>(agg, t, dinv, b3, bn_sum, bn_sq, NN);
    bn_relu_kernel<<<fe_blocks, TB, 0, stream>>>(agg, bn_sum, bn_sq, bn4_g, bn4_b, h, NN);

    // ---- global mean pool ----
    zero_kernel<<<(int)(((long)NG * CH + NG + TB - 1) / TB), TB, 0, stream>>>(pooled, (long)NG * CH + NG);
    pool_accum_kernel<<<fe_blocks, TB, 0, stream>>>(h, batch, pooled, NN);
    pool_count_kernel<<<(NN + TB - 1) / TB, TB, 0, stream>>>(batch, cnt, NN);
    pool_div_kernel<<<(int)(((long)NG * CH + TB - 1) / TB), TB, 0, stream>>>(pooled, cnt, gbuf, NG);

    // ---- MLP head ----
    wmma_gemm_f32<<<gemm_mlp, gemm_blk, 0, stream>>>(gbuf, lin1_W, lin1_b, g1buf, NG, CH, CH, 2);
    wmma_gemm_f32<<<gemm_mlp, gemm_blk, 0, stream>>>(g1buf, lin2_W, lin2_b, out, NG, CH, CH, 1);
}